// DistanceWeightedMessagePassing_52785148067992
// MI455X (gfx1250) — compile-verified
//
#include <hip/hip_runtime.h>
#include <cstdint>
#include <cstddef>

// ---------------------------------------------------------------------------
// GravNet-style DistanceWeightedMessagePassing for MI455X (gfx1250, wave32).
//
// Roofline: ~6 GFLOP total vs ~230 MB mandatory HBM traffic -> memory bound
// (~10us floor @ 23.3 TB/s). The [N,64] feature table (25.6 MB) fits in the
// 192 MB L2, so the 3x819MB KNN gather streams are L2 traffic -> make them
// wide & coalesced (float2/lane, 256B per neighbour row). GEMMs use native
// fp32 WMMA (V_WMMA_F32_16X16X4_F32): full precision, frees the VALU.
// ---------------------------------------------------------------------------

typedef float v2f __attribute__((ext_vector_type(2)));
typedef float v8f __attribute__((ext_vector_type(8)));

#define K_NEIGH    32
#define DOUT       64
#define OUT_STRIDE 448   // 3*128 + 64
#define NEG_FLT_MAX (-3.4028234663852886e38f)

// Per-k-pair LDS stride for the swizzled W: 128 payload floats + pad to 160.
// 160 mod 64 == 32, so the two lane-halves (k-pairs kp and kp+1) cover
// disjoint 32-bank sets -> conflict-free ds_load_b64 B-fragment fetches.
#define KP_STRIDE  160

// ---------------------------------------------------------------------------
// GEMM + bias + ReLU:  H[N,64] = relu(A[N,DIN] @ W[DIN,64] + b)
// One wave32 owns a 16x64 tile: 4 x v8f accumulators, K-loop in steps of 4.
//
// fp32 WMMA operand layout (ISA 7.12.2, 16x4 A / 4x16 B, wave32):
//   A: lane L (m = L&15, half = L>>4): vgpr0 = A[m][k0+2*half],
//                                      vgpr1 = A[m][k0+2*half+1]
//   B: lane L (n = L&15, half = L>>4): vgpr0 = B[k0+2*half][n],
//                                      vgpr1 = B[k0+2*half+1][n]  (mirrors A)
//   C/D: vgpr r, lane L: row = r + 8*half, col = L&15
//
// W is staged in LDS k-pair interleaved: sW[kp*160 + n*2 + r] = W[2kp+r][n],
// so each lane's B operand is a single 8B contiguous ds_load_b64 that lands
// in an even-aligned VGPR pair (no v_mov shuffling before the WMMA).
// ---------------------------------------------------------------------------
template <int DIN>
__global__ __launch_bounds__(64) void gemm_relu_wmma(
    const float* __restrict__ A, int lda,
    const float* __restrict__ W,      // [DIN,64] row-major
    const float* __restrict__ bias,   // [64]
    float* __restrict__ H,            // [N,64]
    int nrows)
{
    __shared__ float sW[(DIN / 2) * KP_STRIDE + DOUT];
    float* sB = sW + (DIN / 2) * KP_STRIDE;

    for (int i = threadIdx.x; i < DIN * DOUT; i += 64) {
        const int k = i >> 6, n = i & 63;
        sW[(k >> 1) * KP_STRIDE + n * 2 + (k & 1)] = W[i];
    }
    if (threadIdx.x < DOUT) sB[threadIdx.x] = bias[threadIdx.x];
    __syncthreads();

    const int lane = threadIdx.x & 31;
    const int wave = threadIdx.x >> 5;        // 2 waves / block
    const int half = lane >> 4;               // K-pair selector
    const int m    = lane & 15;               // row (A) / col (B) within tile
    const int rowBase = (blockIdx.x * 2 + wave) * 16;

    // Clamp loads so EXEC stays all-ones through the WMMAs (ISA requirement).
    const int rA = min(rowBase + m, nrows - 1);
    const float* aRow = A + (size_t)rA * lda;

    v8f acc[4] = {v8f{}, v8f{}, v8f{}, v8f{}};

#pragma unroll
    for (int k0 = 0; k0 < DIN; k0 += 4) {
        const v2f a = *(const v2f*)(aRow + k0 + 2 * half);          // b64
        const float* bp = sW + ((k0 >> 1) + half) * KP_STRIDE + m * 2;
#pragma unroll
        for (int c = 0; c < 4; ++c) {
            const v2f b = *(const v2f*)(bp + c * 32);               // ds b64
            acc[c] = __builtin_amdgcn_wmma_f32_16x16x4_f32(
                false, a, false, b, (short)0, acc[c], false, false);
        }
    }

#pragma unroll
    for (int c = 0; c < 4; ++c) {
        const int col = c * 16 + m;
        const float bv = sB[col];
#pragma unroll
        for (int r = 0; r < 8; ++r) {
            const int row = rowBase + r + half * 8;
            float v = acc[c][r] + bv;
            v = v > 0.0f ? v : 0.0f;
            if (row < nrows) H[(size_t)row * DOUT + col] = v;
        }
    }
}

// ---------------------------------------------------------------------------
// KNN aggregation: for each node, over its 32 neighbours j:
//   wg = exp(-10*d2) * H[j]; write [mean(wg)-prev, max(wg)-prev] into the
// final output at column colOff (these columns double as the next GEMM input).
// 32 threads per node, float2 per thread -> one coalesced 256B burst per
// neighbour row out of L2 with half the VMEM instructions; 8 nodes / block.
// ---------------------------------------------------------------------------
__global__ __launch_bounds__(256) void knn_agg(
    const float* __restrict__ H,     // [N,64]
    const int*   __restrict__ idx,   // [N,32]
    const float* __restrict__ dsq,   // [N,32]
    float* __restrict__ out,         // [N,448]
    int colOff, int nrows)
{
    __shared__ int   sIdx[8][K_NEIGH];
    __shared__ float sWgt[8][K_NEIGH];

    const int g = threadIdx.x >> 5;          // node slot within block
    const int t = threadIdx.x & 31;          // feature pair {2t, 2t+1}
    const int node = blockIdx.x * 8 + g;

    if (node < nrows) {
        sIdx[g][t] = idx[(size_t)node * K_NEIGH + t];
        sWgt[g][t] = __expf(-10.0f * dsq[(size_t)node * K_NEIGH + t]);
    }
    __syncthreads();
    if (node >= nrows) return;

    float s0 = 0.0f, s1 = 0.0f;
    float m0 = NEG_FLT_MAX, m1 = NEG_FLT_MAX;
#pragma unroll 8
    for (int k = 0; k < K_NEIGH; ++k) {
        const v2f hv = *(const v2f*)(H + (size_t)sIdx[g][k] * DOUT + 2 * t);
        const float w = sWgt[g][k];
        const float v0 = hv.x * w, v1 = hv.y * w;
        s0 += v0;             s1 += v1;
        m0 = fmaxf(m0, v0);   m1 = fmaxf(m1, v1);
    }

    const v2f prev = *(const v2f*)(H + (size_t)node * DOUT + 2 * t);
    float* o = out + (size_t)node * OUT_STRIDE + colOff;
    *(v2f*)(o + 2 * t)        = v2f{s0 * (1.0f / K_NEIGH) - prev.x,
                                    s1 * (1.0f / K_NEIGH) - prev.y};
    *(v2f*)(o + DOUT + 2 * t) = v2f{m0 - prev.x, m1 - prev.y};
}

// Passthrough of x into the last 64 output columns (float4 wide).
__global__ __launch_bounds__(256) void copy_x(
    const float* __restrict__ x, float* __restrict__ out, int nrows)
{
    const int i = blockIdx.x * 256 + threadIdx.x;   // index in float4 units
    if (i < nrows * (DOUT / 4)) {
        const int n = i >> 4, f = (i & 15) * 4;
        const float4 v = *(const float4*)(x + (size_t)n * DOUT + f);
        *(float4*)(out + (size_t)n * OUT_STRIDE + 3 * 2 * DOUT + f) = v;
    }
}

extern "C" void kernel_launch(void* const* d_in, const int* in_sizes, int n_in,
                              void* d_out, int out_size, void* d_ws, size_t ws_size,
                              hipStream_t stream)
{
    const float* x   = (const float*)d_in[0];
    const int*   idx = (const int*)  d_in[1];
    const float* dsq = (const float*)d_in[2];
    const float* W0  = (const float*)d_in[3];
    const float* b0  = (const float*)d_in[4];
    const float* W1  = (const float*)d_in[5];
    const float* b1  = (const float*)d_in[6];
    const float* W2  = (const float*)d_in[7];
    const float* b2  = (const float*)d_in[8];
    float* out = (float*)d_out;
    float* h   = (float*)d_ws;                 // reusable [N,64] fp32 buffer

    const int N = in_sizes[0] / DOUT;          // 100000
    const int gemmBlocks = (N + 31) / 32;      // 2 waves x 16 rows per block
    const int aggBlocks  = (N + 7) / 8;        // 8 nodes per 256-thread block

    // Layer 0: x[N,64] -> h -> agg -> out cols [0,128)
    gemm_relu_wmma<64><<<gemmBlocks, 64, 0, stream>>>(x, DOUT, W0, b0, h, N);
    knn_agg<<<aggBlocks, 256, 0, stream>>>(h, idx, dsq, out, 0, N);

    // Layer 1: out cols [0,128) (lda=448) -> h -> agg -> out cols [128,256)
    gemm_relu_wmma<128><<<gemmBlocks, 64, 0, stream>>>(out + 0, OUT_STRIDE, W1, b1, h, N);
    knn_agg<<<aggBlocks, 256, 0, stream>>>(h, idx, dsq, out, 128, N);

    // Layer 2: out cols [128,256) -> h -> agg -> out cols [256,384)
    gemm_relu_wmma<128><<<gemmBlocks, 64, 0, stream>>>(out + 128, OUT_STRIDE, W2, b2, h, N);
    knn_agg<<<aggBlocks, 256, 0, stream>>>(h, idx, dsq, out, 256, N);

    // out cols [384,448) = x
    copy_x<<<(N * (DOUT / 4) + 255) / 256, 256, 0, stream>>>(x, out, N);
}